// P_Model_eff_49855980372255
// MI455X (gfx1250) — compile-verified
//
#include <hip/hip_runtime.h>

// ---------------------------------------------------------------------------
// out = T @ S0 where T (5*ns x 4) = stacked [rows of P^t ; q^T P^t],
// S0 = 4 x N (rows x0,x1,h0,h1). Flat output row m == row m of T @ S0.
// Bandwidth-bound (500 MB streaming stores, ~22 us at 23.3 TB/s):
//   - V_WMMA_F32_16X16X4_F32 per 16x16 output tile (K=4 in one WMMA)
//   - nontemporal b32 stores in SADDR+scale_offset form (1 v_add per store)
//   - NO predicated stores: row/column tails handled by overlapping tiles,
//     which rewrite bitwise-identical values (row m is a pure function of m).
// ---------------------------------------------------------------------------

typedef __attribute__((ext_vector_type(2))) float v2f;
typedef __attribute__((ext_vector_type(8))) float v8f;

// Build T (mpad x 4, row-major, zero-padded) in workspace. Serial, trivial.
__global__ void build_T_kernel(const float* __restrict__ pZM,
                               const float* __restrict__ pZU,
                               const float* __restrict__ pVM,
                               const float* __restrict__ pVU,
                               float* __restrict__ T, int ns, int mpad)
{
    if (threadIdx.x != 0 || blockIdx.x != 0) return;
    const float zm = *pZM, zu = *pZU, vm = *pVM, vu = *pVU;
    float P[4][4] = {{1.f, 1.f, 0.f, 0.f},
                     {0.f, 1.f, zm,  zu },
                     {1.f, 1.f, 0.f, 0.f},
                     {0.f, 0.f, vm,  vu }};
    float A[4][4] = {{1.f,0.f,0.f,0.f},
                     {0.f,1.f,0.f,0.f},
                     {0.f,0.f,1.f,0.f},
                     {0.f,0.f,0.f,1.f}};  // A = P^0
    for (int t = 0; t < ns; ++t) {
        for (int k = 0; k < 4; ++k)
            for (int j = 0; j < 4; ++j)
                T[(t*4 + k)*4 + j] = A[k][j];               // traj rows: P^t
        for (int j = 0; j < 4; ++j)
            T[(4*ns + t)*4 + j] = zm*A[2][j] + zu*A[3][j];  // con row: q^T P^t
        float Bm[4][4];
        for (int k = 0; k < 4; ++k)
            for (int j = 0; j < 4; ++j) {
                float s = 0.f;
                for (int i = 0; i < 4; ++i) s += P[k][i] * A[i][j];
                Bm[k][j] = s;
            }
        for (int k = 0; k < 4; ++k)
            for (int j = 0; j < 4; ++j) A[k][j] = Bm[k][j];
    }
    for (int r = 5*ns; r < mpad; ++r)                       // zero padding rows
        for (int j = 0; j < 4; ++j) T[r*4 + j] = 0.f;
}

__device__ __forceinline__ v8f wmma_16x16x4(v2f a, v2f b) {
    v8f c = {};
    return __builtin_amdgcn_wmma_f32_16x16x4_f32(false, a, false, b,
                                                 (short)0, c, false, false);
}

// Specialized: MT row tiles, A fragments resident in registers (2*MT VGPRs),
// last row tile overlapped onto rows [M_rows-16, M_rows) -> no store guards.
// Requires M_rows in (16*(MT-1), 16*MT] and N >= 16.
template<int MT>
__global__ __launch_bounds__(256)
void gemm_T_fixed(const float* __restrict__ x0, const float* __restrict__ x1,
                  const float* __restrict__ h0, const float* __restrict__ h1,
                  const float* __restrict__ T,  float* __restrict__ out,
                  int N, int n_tiles, int M_rows)
{
    const int  lane   = threadIdx.x & 31;
    const int  l15    = lane & 15;
    const bool hi     = lane >= 16;                   // lane half selects K-pair
    const int  gwave  = blockIdx.x * (blockDim.x >> 5) + (threadIdx.x >> 5);
    const int  nwaves = gridDim.x * (blockDim.x >> 5);
    const unsigned uN = (unsigned)N;

    // A tiles: 16x4 f32; lane (M=l15) holds K=(0,1) low half / K=(2,3) high half.
    // Last tile overlaps: rows M_rows-16 .. M_rows-1 (duplicate rows rewrite
    // identical values -> no guards needed anywhere).
    v2f a[MT];
    #pragma unroll
    for (int mt = 0; mt < MT; ++mt) {
        const int rb = (mt == MT - 1) ? (M_rows - 16) : mt * 16;
        a[mt] = *(const v2f*)(T + (rb + l15)*4 + (hi ? 2 : 0));
    }
    const unsigned lastbase = (unsigned)(M_rows - 16) * uN;  // SGPR, hoisted

    // B rows: lanes 0-15 hold K=0,1 (x0,x1); lanes 16-31 hold K=2,3 (h0,h1).
    const float* __restrict__ bsrc0 = hi ? h0 : x0;
    const float* __restrict__ bsrc1 = hi ? h1 : x1;

    for (int ct = gwave; ct < n_tiles; ct += nwaves) {  // wave-uniform loop
        int n0 = ct * 16;
        if (n0 > N - 16) n0 = N - 16;                   // column overlap (uniform)
        const int n = n0 + l15;

        v2f b; b.x = bsrc0[n]; b.y = bsrc1[n];

        const unsigned ubase = (hi ? 8u * uN : 0u) + (unsigned)n;

        #pragma unroll
        for (int mt = 0; mt < MT; ++mt) {
            const v8f d = wmma_16x16x4(a[mt], b);
            // D layout: lanes 0-15 -> M=row_base+j ; lanes 16-31 -> M=row_base+8+j
            unsigned off = ((mt == MT - 1) ? lastbase
                                           : (unsigned)(mt * 16) * uN) + ubase;
            #pragma unroll
            for (int j = 0; j < 8; ++j) {
                __builtin_nontemporal_store(d[j], out + off);  // unconditional
                off += uN;
            }
        }
    }
}

// Generic fallback (any ns / tiny N): runtime row-tile count, guarded stores.
__global__ __launch_bounds__(256)
void gemm_T_generic(const float* __restrict__ x0, const float* __restrict__ x1,
                    const float* __restrict__ h0, const float* __restrict__ h1,
                    const float* __restrict__ T,  float* __restrict__ out,
                    int N, int n_tiles, int m_tiles, int M_rows)
{
    const int  lane   = threadIdx.x & 31;
    const int  l15    = lane & 15;
    const bool hi     = lane >= 16;
    const int  gwave  = blockIdx.x * (blockDim.x >> 5) + (threadIdx.x >> 5);
    const int  nwaves = gridDim.x * (blockDim.x >> 5);
    const unsigned uN = (unsigned)N;

    const float* __restrict__ bsrc0 = hi ? h0 : x0;
    const float* __restrict__ bsrc1 = hi ? h1 : x1;

    for (int ct = gwave; ct < n_tiles; ct += nwaves) {
        const int n0 = ct * 16;
        const int n  = n0 + l15;
        const int nc = n < N ? n : (N - 1);
        const bool nvalid = n < N;
        v2f b; b.x = bsrc0[nc]; b.y = bsrc1[nc];

        unsigned off0 = (hi ? 8u * uN : 0u) + (unsigned)n;
        for (int mt = 0; mt < m_tiles; ++mt) {
            const v2f a = *(const v2f*)(T + (mt*16 + l15)*4 + (hi ? 2 : 0));
            const v8f d = wmma_16x16x4(a, b);
            unsigned off = off0;
            #pragma unroll
            for (int j = 0; j < 8; ++j) {
                const int m = mt*16 + (hi ? 8 : 0) + j;
                if (nvalid && (m < M_rows))
                    __builtin_nontemporal_store(d[j], out + off);
                off += uN;
            }
            off0 += 16u * uN;
        }
    }
}

extern "C" void kernel_launch(void* const* d_in, const int* in_sizes, int n_in,
                              void* d_out, int out_size, void* d_ws, size_t ws_size,
                              hipStream_t stream) {
    const float* x0 = (const float*)d_in[0];
    const float* x1 = (const float*)d_in[1];
    const float* h0 = (const float*)d_in[2];
    const float* h1 = (const float*)d_in[3];
    const float* ZM = (const float*)d_in[4];
    const float* ZU = (const float*)d_in[5];
    const float* VM = (const float*)d_in[6];
    const float* VU = (const float*)d_in[7];

    // num_steps lives on device; recover from output geometry: out = 5*ns*N.
    const int N  = in_sizes[0];
    const int ns = out_size / (5 * N);          // = 50
    const int M_rows  = 5 * ns;                 // = 250 flat output rows
    const int m_tiles = (M_rows + 15) / 16;     // = 16
    const int mpad    = m_tiles * 16;           // = 256 (T zero-padded)

    float* T = (float*)d_ws;                    // mpad*4 floats = 4 KB
    build_T_kernel<<<1, 1, 0, stream>>>(ZM, ZU, VM, VU, T, ns, mpad);

    const int n_tiles = (N + 15) / 16;          // = 31250
    const int waves_per_block = 256 / 32;
    const int blocks = (n_tiles + waves_per_block - 1) / waves_per_block;

    if (m_tiles == 16 && N >= 16) {
        gemm_T_fixed<16><<<blocks, 256, 0, stream>>>(
            x0, x1, h0, h1, T, (float*)d_out, N, n_tiles, M_rows);
    } else {
        gemm_T_generic<<<blocks, 256, 0, stream>>>(
            x0, x1, h0, h1, T, (float*)d_out, N, n_tiles, m_tiles, M_rows);
    }
}